// _PartitionedProductLayer_60206851555328
// MI455X (gfx1250) — compile-verified
//
#include <hip/hip_runtime.h>

typedef __attribute__((ext_vector_type(2))) float v2f;
typedef __attribute__((ext_vector_type(8))) float v8f;

// Parity sign of the geometric product of basis blades a,b (Cl(6,0,0): metric all +1)
__device__ __forceinline__ float cayley_sign(int a, int b) {
    int cnt = 0;
    int aa = a >> 1;
    while (aa) { cnt += __popc(aa & b); aa >>= 1; }
    return (cnt & 1) ? -1.0f : 1.0f;
}

// out[b,k] = sum_i x[b,i] * M[i,k],  M[i,k] = sign(i, i^k) * weight[i^k]
// One wave per 16-row x 64-col output tile; WMMA f32 16x16x4, K=64 in 16 steps.
__global__ __launch_bounds__(256) void geoprod_wmma_kernel(
    const float* __restrict__ x, const float* __restrict__ weight,
    float* __restrict__ out, int nrows)
{
    // M (64x64) pre-swizzled into WMMA B-fragment order:
    // fragment f = s*4 + nb occupies floats [f*64, f*64+64):
    //   slot (lane L, vgpr v) at f*64 + L*2 + v holds
    //   M[K][N], K = 4s + (L<16 ? v : v+2), N = nb*16 + (L&15)
    __shared__ float ldsB[64 * 64];

    const int tid = threadIdx.x;

    // ---- Phase 1: build B fragments in LDS ----
    for (int idx = tid; idx < 4096; idx += 256) {
        int f = idx >> 6;
        int r = idx & 63;
        int L = r >> 1;
        int v = r & 1;
        int s  = f >> 2;
        int nb = f & 3;
        int K = 4 * s + ((L < 16) ? v : v + 2);
        int N = nb * 16 + (L & 15);
        int j = K ^ N;                      // source blade index in weight
        ldsB[idx] = cayley_sign(K, j) * weight[j];
    }
    __syncthreads();

    // ---- Phase 2: stream x through the matrix pipe ----
    const int wave = tid >> 5;
    const int lane = tid & 31;
    const int tileIdx = blockIdx.x * 8 + wave;
    const int row0 = tileIdx * 16;
    if (row0 >= nrows) return;

    const int mrow  = lane & 15;          // A row (and N column for B/D) handled by this lane
    const int khalf = (lane >> 4) * 2;    // lanes 16-31 hold the upper 2 K-values of each step

    // A fragments: a[s].x -> VGPR0 (K = 4s+khalf), a[s].y -> VGPR1 (K = 4s+khalf+1)
    const float* xrow = x + (size_t)(row0 + mrow) * 64 + khalf;
    v2f a[16];
    #pragma unroll
    for (int s = 0; s < 16; ++s) {
        a[s] = *(const v2f*)(xrow + 4 * s);     // global_load_b64, coalesced in row-pairs
    }

    v8f acc0 = {}, acc1 = {}, acc2 = {}, acc3 = {};
    #pragma unroll
    for (int s = 0; s < 16; ++s) {
        const v2f* bp = (const v2f*)(ldsB + s * 256) + lane; // ds_load_b64 per fragment
        v2f b0 = bp[0];
        v2f b1 = bp[32];
        v2f b2 = bp[64];
        v2f b3 = bp[96];
        acc0 = __builtin_amdgcn_wmma_f32_16x16x4_f32(false, a[s], false, b0, (short)0, acc0, false, false);
        acc1 = __builtin_amdgcn_wmma_f32_16x16x4_f32(false, a[s], false, b1, (short)0, acc1, false, false);
        acc2 = __builtin_amdgcn_wmma_f32_16x16x4_f32(false, a[s], false, b2, (short)0, acc2, false, false);
        acc3 = __builtin_amdgcn_wmma_f32_16x16x4_f32(false, a[s], false, b3, (short)0, acc3, false, false);
    }

    // ---- Store D tiles: VGPR v -> row v (lanes 0-15) / row v+8 (lanes 16-31) ----
    const int rbase = row0 + ((lane >> 4) ? 8 : 0);
    float* orow = out + (size_t)rbase * 64 + (lane & 15);
    #pragma unroll
    for (int v = 0; v < 8; ++v) {
        size_t ro = (size_t)v * 64;
        orow[ro +  0] = acc0[v];
        orow[ro + 16] = acc1[v];
        orow[ro + 32] = acc2[v];
        orow[ro + 48] = acc3[v];
    }
}

extern "C" void kernel_launch(void* const* d_in, const int* in_sizes, int n_in,
                              void* d_out, int out_size, void* d_ws, size_t ws_size,
                              hipStream_t stream) {
    const float* x      = (const float*)d_in[0];
    const float* weight = (const float*)d_in[1];
    float* out          = (float*)d_out;

    int nrows  = in_sizes[0] / 64;            // 1048576
    int tiles  = (nrows + 15) / 16;           // 16 rows per wave-tile
    int blocks = (tiles + 7) / 8;             // 8 waves per 256-thread block

    geoprod_wmma_kernel<<<blocks, 256, 0, stream>>>(x, weight, out, nrows);
}